// SequenceTrimmer_50319836840059
// MI455X (gfx1250) — compile-verified
//
#include <hip/hip_runtime.h>
#include <cstdint>

// ---------------------------------------------------------------------------
// SequenceTrimmer for MI455X (gfx1250).
// Pure data-movement problem: ~530 MB of traffic, ~23 us at 23.3 TB/s HBM.
// CDNA5 feature used: GLOBAL_LOAD_ASYNC_TO_LDS_B128 (ASYNCcnt) to stage
// permuted 2KB rows of `uu` into LDS, double-buffered with s_wait_asynccnt,
// so both the global reads and the global writes stay fully coalesced and the
// random j-gather happens in LDS.
// ---------------------------------------------------------------------------

__device__ __forceinline__ unsigned hash32(unsigned x) {
    x ^= x >> 16; x *= 0x7feb352dU;
    x ^= x >> 15; x *= 0x846ca68bU;
    x ^= x >> 16;
    return x;
}

// ---------------------------------------------------------------------------
// Kernel 1: per-sample permutation + valid count.
// perm[b] = [invalid indices in random order] ++ [valid indices, stable order]
// via ascending bitonic sort of 64-bit keys (orderKey << 32 | index).
// ---------------------------------------------------------------------------
#define LP2 512   // L rounded to pow2 (harness: L == 512)

__global__ __launch_bounds__(256) void perm_kernel(
    const unsigned char* __restrict__ mask,   // (B, 1, L) bool (1 byte/elem)
    int* __restrict__ perm,                   // (B, L)
    int* __restrict__ nvalid,                 // (B,)
    int L)
{
    __shared__ unsigned long long key[LP2];
    __shared__ int cnt;

    const int b = blockIdx.x;
    const int t = threadIdx.x;
    const unsigned char* m = mask + (size_t)b * L;

    if (t == 0) cnt = 0;
    __syncthreads();

    int local = 0;
    for (int i = t; i < L; i += blockDim.x) {
        const bool vld = (m[i] != 0);
        local += vld ? 1 : 0;
        unsigned ok;
        if (vld) {
            ok = 0xFFFFFFFFu;                      // valid: last, stable by index
        } else {
            ok = hash32((unsigned)(b * 0x9E3779B9u) ^ (unsigned)i);
            if (ok == 0xFFFFFFFFu) ok = 0xFFFFFFFEu;
        }
        key[i] = ((unsigned long long)ok << 32) | (unsigned)i;
    }
    for (int i = L + t; i < LP2; i += blockDim.x)
        key[i] = ~0ull;                            // padding sorts to the end
    atomicAdd(&cnt, local);
    __syncthreads();

    // Bitonic sort, ascending, LP2 elements with LP2/2 compare-exchanges/stage.
    for (int k = 2; k <= LP2; k <<= 1) {
        for (int j = k >> 1; j >= 1; j >>= 1) {
            for (int tt = t; tt < (LP2 >> 1); tt += blockDim.x) {
                const int i   = ((tt & ~(j - 1)) << 1) | (tt & (j - 1));
                const int ixj = i | j;
                const bool up = ((i & k) == 0);
                const unsigned long long a = key[i];
                const unsigned long long c = key[ixj];
                if (up ? (a > c) : (a < c)) { key[i] = c; key[ixj] = a; }
            }
            __syncthreads();
        }
    }

    for (int i = t; i < L; i += blockDim.x)
        perm[(size_t)b * L + i] = (int)(key[i] & 0xFFFFu);
    if (t == 0) nvalid[b] = cnt;
}

// ---------------------------------------------------------------------------
// Kernel 2: permuted gather + zero-fill for x_p, v_p, mask_p (small arrays).
// Flattened over B * (C + 4 + 1) * L elements.
// ---------------------------------------------------------------------------
__global__ __launch_bounds__(256) void gather_small(
    const float* __restrict__ x, const float* __restrict__ v,
    const unsigned char* __restrict__ mask,
    const int* __restrict__ perm, const int* __restrict__ nvalid,
    float* __restrict__ out_x, float* __restrict__ out_v,
    float* __restrict__ out_m,
    int B, int C, int L)
{
    const long long idx   = (long long)blockIdx.x * blockDim.x + threadIdx.x;
    const long long total = (long long)B * (C + 5) * L;
    if (idx >= total) return;

    const int i  = (int)(idx % L);
    const long long r = idx / L;
    const int ch = (int)(r % (C + 5));
    const int b  = (int)(r / (C + 5));

    const int  pi   = perm[(size_t)b * L + i];
    const bool keep = (i < nvalid[b]);

    if (ch < C) {
        out_x[((size_t)b * C + ch) * L + i] =
            keep ? x[((size_t)b * C + ch) * L + pi] : 0.0f;
    } else if (ch < C + 4) {
        const int cv = ch - C;
        out_v[((size_t)b * 4 + cv) * L + i] =
            keep ? v[((size_t)b * 4 + cv) * L + pi] : 0.0f;
    } else {
        out_m[(size_t)b * L + i] =
            keep ? (mask[(size_t)b * L + pi] ? 1.0f : 0.0f) : 0.0f;
    }
}

// ---------------------------------------------------------------------------
// Kernel 3: uu_p[b,c,i,j] = uu[b,c,perm[i],perm[j]]  (256 MiB each way).
// Per wave: async-DMA row perm[i] (2 KB) into LDS (4 x b128 per lane),
// double-buffered with s_wait_asynccnt, then LDS-gather by perm[j] and emit
// coalesced float4 stores. Each input row is read exactly once.
// ---------------------------------------------------------------------------
#define UU_WAVES       8
#define ROWS_PER_BLOCK 64     // rows handled per block -> 8 rows per wave
#define UU_L           512

__global__ __launch_bounds__(256) void gather_uu(
    const float* __restrict__ uu, const int* __restrict__ perm,
    float* __restrict__ out, int B, int Cu, int L)
{
    __shared__ __align__(16) float rowbuf[UU_WAVES][2][UU_L];
    __shared__ int spermJ[UU_L];

    const int rowGroups = L / ROWS_PER_BLOCK;
    const int bc = blockIdx.x / rowGroups;          // linear (b, c)
    const int rg = blockIdx.x % rowGroups;
    if (bc >= B * Cu) return;
    const int b = bc / Cu;

    const int* permb = perm + (size_t)b * L;
    for (int i = threadIdx.x; i < L; i += blockDim.x) spermJ[i] = permb[i];
    __syncthreads();

    const int wave = threadIdx.x >> 5;              // wave32
    const int lane = threadIdx.x & 31;
    const int rowsPerWave = ROWS_PER_BLOCK / UU_WAVES;   // 8
    const int rowBase = rg * ROWS_PER_BLOCK + wave * rowsPerWave;

    const float* uubase  = uu  + (size_t)bc * L * L;
    float*       outbase = out + (size_t)bc * L * L;

    // Issue one 2KB row load into LDS buffer `bufsel` (4 async b128 per lane).
    auto issue = [&](int r, int bufsel) {
        const int pi = spermJ[rowBase + r];
        const float* src = uubase + (size_t)pi * L;
        const uint32_t ldsbase =
            (uint32_t)(uintptr_t)(&rowbuf[wave][bufsel][0]);
#pragma unroll
        for (int k = 0; k < 4; ++k) {
            const int fidx = (lane + 32 * k) * 4;        // float index, 16B step
            asm volatile("global_load_async_to_lds_b128 %0, %1, off"
                         :: "v"(ldsbase + (uint32_t)fidx * 4u),
                            "v"(src + fidx)
                         : "memory");
        }
    };

    // Gather row from LDS by perm[j], store coalesced float4s.
    auto process = [&](int r, int bufsel) {
        const float* buf = &rowbuf[wave][bufsel][0];
        float* dstrow = outbase + (size_t)(rowBase + r) * L;
#pragma unroll
        for (int k = 0; k < 4; ++k) {
            const int j0 = 4 * (lane + 32 * k);
            float4 val;
            val.x = buf[spermJ[j0 + 0]];
            val.y = buf[spermJ[j0 + 1]];
            val.z = buf[spermJ[j0 + 2]];
            val.w = buf[spermJ[j0 + 3]];
            *(float4*)(dstrow + j0) = val;
        }
    };

    issue(0, 0);
    for (int r = 0; r < rowsPerWave; ++r) {
        if (r + 1 < rowsPerWave) {
            issue(r + 1, (r + 1) & 1);
            // 4 loads of row r+1 outstanding; row r's 4 loads complete.
            asm volatile("s_wait_asynccnt 4" ::: "memory");
        } else {
            asm volatile("s_wait_asynccnt 0" ::: "memory");
        }
        process(r, r & 1);
    }
}

// ---------------------------------------------------------------------------
// Host launch.
// Inputs (setup_inputs order): x (B,C,L) f32, v (B,4,L) f32,
// uu (B,Cu,L,L) f32, mask (B,1,L) bool (1 byte/elem).
// Output: concat of x_p, v_p, mask_p (as 0/1 float), uu_p.
// ---------------------------------------------------------------------------
extern "C" void kernel_launch(void* const* d_in, const int* in_sizes, int n_in,
                              void* d_out, int out_size, void* d_ws, size_t ws_size,
                              hipStream_t stream)
{
    const float*         x    = (const float*)d_in[0];
    const float*         v    = (const float*)d_in[1];
    const float*         uu   = (const float*)d_in[2];
    const unsigned char* mask = (const unsigned char*)d_in[3];

    const long long sx = in_sizes[0];   // B*C*L
    const long long sv = in_sizes[1];   // B*4*L
    const long long su = in_sizes[2];   // B*Cu*L*L
    const long long sm = in_sizes[3];   // B*L

    // Cu == 4 in this problem => su/sv == L; then B, C, Cu follow.
    const int L  = (int)(su / sv);
    const int B  = (int)(sm / L);
    const int C  = (int)(sx / ((long long)B * L));
    const int Cu = (int)(su / ((long long)B * L * L));

    // Workspace: perm (B*L int) + nvalid (B int)
    int* perm   = (int*)d_ws;
    int* nvalid = perm + (size_t)B * L;

    float* out_x  = (float*)d_out;
    float* out_v  = out_x + (size_t)B * C * L;
    float* out_m  = out_v + (size_t)B * 4 * L;
    float* out_uu = out_m + (size_t)B * L;

    // 1) permutation + valid counts (one block per sample)
    perm_kernel<<<B, LP2 / 2, 0, stream>>>(mask, perm, nvalid, L);

    // 2) small gathers (x, v, mask) with zero-fill
    {
        const long long total = (long long)B * (C + 5) * L;
        const int blocks = (int)((total + 255) / 256);
        gather_small<<<blocks, 256, 0, stream>>>(x, v, mask, perm, nvalid,
                                                 out_x, out_v, out_m, B, C, L);
    }

    // 3) uu double-permutation via async-LDS staging
    {
        const int blocks = (B * Cu) * (L / ROWS_PER_BLOCK);
        gather_uu<<<blocks, 256, 0, stream>>>(uu, perm, out_uu, B, Cu, L);
    }
}